// GraphGNN_25872882991822
// MI455X (gfx1250) — compile-verified
//
#include <hip/hip_runtime.h>

#define N_NODES 50000
#define N_EDGES 1600000
#define N_FEAT  50
#define K1P     52          // N_FEAT padded to multiple of 4
#define HID     20
#define OUT_DIM 10
#define N_GRAPHS 64

typedef __attribute__((ext_vector_type(2))) float v2f;
typedef __attribute__((ext_vector_type(8))) float v8f;

// ---------------- utility kernels ----------------

__global__ void zero_f32(float* __restrict__ p, long long n) {
    long long t = (long long)blockIdx.x * blockDim.x + threadIdx.x;
    if (t < n) p[t] = 0.0f;
}

// x [N_NODES, N_FEAT] -> xp [N_NODES, K1P], zero-padded tail.
__global__ void pad_x_kernel(const float* __restrict__ x, float* __restrict__ xp) {
    long long t = (long long)blockIdx.x * blockDim.x + threadIdx.x;
    if (t >= (long long)N_NODES * K1P) return;
    int node = (int)(t / K1P);
    int f    = (int)(t - (long long)node * K1P);
    xp[t] = (f < N_FEAT) ? x[(long long)node * N_FEAT + f] : 0.0f;
}

// W [K, HID] -> Wp pair-interleaved: Wp[(k>>1)*64 + n*2 + (k&1)], size Kp*32 floats.
__global__ void pad_w_kernel(const float* __restrict__ W, float* __restrict__ Wp,
                             int K, int Kp) {
    int t = blockIdx.x * blockDim.x + threadIdx.x;
    if (t >= Kp * 32) return;
    int kk = t >> 6, r = t & 63;
    int n = r >> 1, half = r & 1;
    int k = kk * 2 + half;
    Wp[t] = (k < K && n < HID) ? W[k * HID + n] : 0.0f;
}

// ---------------- edge aggregation (scatter-add) ----------------
// reads src rows with `stride`, writes dst rows with `stride`; features f < feat.

__global__ void scatter_add_kernel(const float* __restrict__ x,
                                   const int* __restrict__ src,
                                   const int* __restrict__ dst,
                                   float* __restrict__ agg, int feat, int stride) {
    long long t = (long long)blockIdx.x * blockDim.x + threadIdx.x;
    long long total = (long long)N_EDGES * feat;
    if (t >= total) return;
    int e = (int)(t / feat);
    int f = (int)(t - (long long)e * feat);
    float v = x[(long long)src[e] * stride + f];
    atomicAdd(&agg[(long long)dst[e] * stride + f], v);
}

// ---------------- fused GraphConv GEMM: relu(A1@W1 + A2@W2 + b) ----------------
// A1, A2: [N_NODES, Kp] row-major, K-padded with zeros.  Wp1, Wp2: pair-interleaved
// [Kp*32].  out: [N_NODES, HID].  One wave per 16-node tile; two 16x16 N-tiles.

__global__ void gemm_relu_kernel(const float* __restrict__ A1,
                                 const float* __restrict__ A2,
                                 const float* __restrict__ Wp1,
                                 const float* __restrict__ Wp2,
                                 const float* __restrict__ bias,
                                 float* __restrict__ out,
                                 int Kp) {
    __shared__ __align__(16) float sW[2 * K1P * 32];   // max layer-1 footprint (13.3 KB)

    // cooperative LDS fill of both weight matrices (all threads participate)
    const int wtot = Kp * 32;
    for (int i = threadIdx.x; i < 2 * wtot; i += blockDim.x)
        sW[i] = (i < wtot) ? Wp1[i] : Wp2[i - wtot];
    __syncthreads();

    int wave = (int)((blockIdx.x * (long long)blockDim.x + threadIdx.x) >> 5);
    int lane = threadIdx.x & 31;
    int m0 = wave * 16;
    if (m0 >= N_NODES) return;                 // wave-uniform: EXEC stays all-ones

    const int mrow = m0 + (lane & 15);         // A row for this lane
    const int koff = (lane >> 4) * 2;          // lane half selects K {0,1} vs {2,3}
    const int nl   = lane & 15;                // B/C column within tile

    const v2f* __restrict__ arow1 = (const v2f*)(A1 + (long long)mrow * Kp);
    const v2f* __restrict__ arow2 = (const v2f*)(A2 + (long long)mrow * Kp);
    const v2f* __restrict__ sW1 = (const v2f*)sW;
    const v2f* __restrict__ sW2 = (const v2f*)(sW + wtot);

    v8f acc0 = {}; v8f acc1 = {};

    for (int k = 0; k < Kp; k += 4) {
        int kh = (k + koff) >> 1;              // pair index into A rows / W pairs
        v2f a1 = arow1[kh];
        v2f a2 = arow2[kh];
        v2f b10 = sW1[kh * 32 + nl];
        v2f b11 = sW1[kh * 32 + nl + 16];
        v2f b20 = sW2[kh * 32 + nl];
        v2f b21 = sW2[kh * 32 + nl + 16];
        acc0 = __builtin_amdgcn_wmma_f32_16x16x4_f32(false, a1, false, b10, (short)0, acc0, false, false);
        acc1 = __builtin_amdgcn_wmma_f32_16x16x4_f32(false, a1, false, b11, (short)0, acc1, false, false);
        acc0 = __builtin_amdgcn_wmma_f32_16x16x4_f32(false, a2, false, b20, (short)0, acc0, false, false);
        acc1 = __builtin_amdgcn_wmma_f32_16x16x4_f32(false, a2, false, b21, (short)0, acc1, false, false);
    }

    // ---- bias + relu + store (C layout: VGPR j -> rows j / j+8 across lane halves)
    int rbase = m0 + ((lane >> 4) << 3);
    #pragma unroll
    for (int j = 0; j < 8; j++) {
        int row = rbase + j;
        float v0 = acc0[j] + bias[nl];
        out[(long long)row * HID + nl] = fmaxf(v0, 0.0f);
        int c1 = 16 + nl;
        if (c1 < HID) {
            float v1 = acc1[j] + bias[c1];
            out[(long long)row * HID + c1] = fmaxf(v1, 0.0f);
        }
    }
}

// ---------------- pooling: per-graph max / sum / count ----------------
// h is post-ReLU (>=0) so integer atomicMax on float bits with 0-init is exact.

__global__ void pool_kernel(const float* __restrict__ h, const int* __restrict__ batch,
                            float* __restrict__ maxP, float* __restrict__ sumP,
                            float* __restrict__ cnt) {
    int t = blockIdx.x * blockDim.x + threadIdx.x;
    if (t >= N_NODES * HID) return;
    int node = t / HID, f = t - node * HID;
    int g = batch[node];
    float v = h[t];
    atomicMax((int*)&maxP[g * HID + f], __float_as_int(v));
    atomicAdd(&sumP[g * HID + f], v);
    if (f == 0) atomicAdd(&cnt[g], 1.0f);
}

// ---------------- final linear on cat([max, mean]) ----------------

__global__ void final_lin_kernel(const float* __restrict__ maxP,
                                 const float* __restrict__ sumP,
                                 const float* __restrict__ cnt,
                                 const float* __restrict__ W_lin,
                                 const float* __restrict__ b_lin,
                                 float* __restrict__ out) {
    int t = blockIdx.x * blockDim.x + threadIdx.x;
    if (t >= N_GRAPHS * OUT_DIM) return;
    int g = t / OUT_DIM, o = t - g * OUT_DIM;
    float c = fmaxf(cnt[g], 1.0f);
    float acc = b_lin[o];
    #pragma unroll
    for (int hh = 0; hh < HID; hh++) {
        acc += maxP[g * HID + hh] * W_lin[hh * OUT_DIM + o];
        acc += (sumP[g * HID + hh] / c) * W_lin[(HID + hh) * OUT_DIM + o];
    }
    out[t] = acc;
}

// ---------------- host orchestration ----------------

static inline long long cdivll(long long a, long long b) { return (a + b - 1) / b; }

extern "C" void kernel_launch(void* const* d_in, const int* in_sizes, int n_in,
                              void* d_out, int out_size, void* d_ws, size_t ws_size,
                              hipStream_t stream) {
    const float* x      = (const float*)d_in[0];
    const int*   eidx   = (const int*)  d_in[1];
    const int*   batch  = (const int*)  d_in[2];
    const float* W_rel1 = (const float*)d_in[3];
    const float* W_root1= (const float*)d_in[4];
    const float* b1     = (const float*)d_in[5];
    const float* W_rel2 = (const float*)d_in[6];
    const float* W_root2= (const float*)d_in[7];
    const float* b2     = (const float*)d_in[8];
    const float* W_rel3 = (const float*)d_in[9];
    const float* W_root3= (const float*)d_in[10];
    const float* b3     = (const float*)d_in[11];
    const float* W_lin  = (const float*)d_in[12];
    const float* b_lin  = (const float*)d_in[13];
    float* out = (float*)d_out;

    const int* src = eidx;
    const int* dst = eidx + N_EDGES;

    // workspace layout (floats)
    float* ws = (float*)d_ws;
    const long long PAD_SZ = (long long)N_NODES * K1P;      // 2.6M
    const long long HSZ    = (long long)N_NODES * HID;      // 1.0M
    float* agg  = ws;                                       // [2.6M] stride 52 L1, stride 20 L2/3
    float* xp   = agg + PAD_SZ;                             // [2.6M] padded x
    float* hA   = xp + PAD_SZ;                              // [1M] h1, later h3
    float* hB   = hA + HSZ;                                 // [1M] h2
    float* wp1r = hB + HSZ;                                 // 52*32
    float* wp1o = wp1r + K1P * 32;
    float* wp2r = wp1o + K1P * 32;                          // 20*32
    float* wp2o = wp2r + HID * 32;
    float* wp3r = wp2o + HID * 32;
    float* wp3o = wp3r + HID * 32;
    float* maxP = wp3o + HID * 32;                          // 64*20
    float* sumP = maxP + N_GRAPHS * HID;                    // 64*20
    float* cnt  = sumP + N_GRAPHS * HID;                    // 64

    const int B = 256;

    // pad inputs/weights
    pad_x_kernel<<<(int)cdivll(PAD_SZ, B), B, 0, stream>>>(x, xp);
    pad_w_kernel<<<(K1P*32 + B - 1)/B, B, 0, stream>>>(W_rel1,  wp1r, N_FEAT, K1P);
    pad_w_kernel<<<(K1P*32 + B - 1)/B, B, 0, stream>>>(W_root1, wp1o, N_FEAT, K1P);
    pad_w_kernel<<<(HID*32 + B - 1)/B, B, 0, stream>>>(W_rel2,  wp2r, HID, HID);
    pad_w_kernel<<<(HID*32 + B - 1)/B, B, 0, stream>>>(W_root2, wp2o, HID, HID);
    pad_w_kernel<<<(HID*32 + B - 1)/B, B, 0, stream>>>(W_rel3,  wp3r, HID, HID);
    pad_w_kernel<<<(HID*32 + B - 1)/B, B, 0, stream>>>(W_root3, wp3o, HID, HID);

    const int gemm_blocks = (int)cdivll(N_NODES / 16, 8);   // 8 waves/block, 1 tile/wave

    // ---- layer 1 (stride-52 padded agg & x) ----
    zero_f32<<<(int)cdivll(PAD_SZ, B), B, 0, stream>>>(agg, PAD_SZ);
    scatter_add_kernel<<<(int)cdivll((long long)N_EDGES * N_FEAT, B), B, 0, stream>>>(xp, src, dst, agg, N_FEAT, K1P);
    gemm_relu_kernel<<<gemm_blocks, B, 0, stream>>>(agg, xp, wp1r, wp1o, b1, hA, K1P);

    // ---- layer 2 ----
    zero_f32<<<(int)cdivll(HSZ, B), B, 0, stream>>>(agg, HSZ);
    scatter_add_kernel<<<(int)cdivll((long long)N_EDGES * HID, B), B, 0, stream>>>(hA, src, dst, agg, HID, HID);
    gemm_relu_kernel<<<gemm_blocks, B, 0, stream>>>(agg, hA, wp2r, wp2o, b2, hB, HID);

    // ---- layer 3 (h3 -> hA) ----
    zero_f32<<<(int)cdivll(HSZ, B), B, 0, stream>>>(agg, HSZ);
    scatter_add_kernel<<<(int)cdivll((long long)N_EDGES * HID, B), B, 0, stream>>>(hB, src, dst, agg, HID, HID);
    gemm_relu_kernel<<<gemm_blocks, B, 0, stream>>>(agg, hB, wp3r, wp3o, b3, hA, HID);

    // ---- pooling ----
    const long long pool_state = 2LL * N_GRAPHS * HID + N_GRAPHS;
    zero_f32<<<(int)cdivll(pool_state, B), B, 0, stream>>>(maxP, pool_state);
    pool_kernel<<<(int)cdivll((long long)N_NODES * HID, B), B, 0, stream>>>(hA, batch, maxP, sumP, cnt);

    // ---- final linear ----
    final_lin_kernel<<<(N_GRAPHS * OUT_DIM + B - 1)/B, B, 0, stream>>>(maxP, sumP, cnt, W_lin, b_lin, out);
}